// AutoRegressiveATT_46110768890424
// MI455X (gfx1250) — compile-verified
//
#include <hip/hip_runtime.h>
#include <hip/hip_bf16.h>

// ---------------------------------------------------------------------------
// AutoRegressiveATT for MI455X (gfx1250), fp32 end-to-end.
//
// Algebraic restructure: pool ctx over q BEFORE the Wo projection (linearity),
// so the dominant work is two fp32 NT-GEMMs done with V_WMMA_F32_16X16X4_F32:
//   GEMM1: qkv  = xt(384x4096)  @ Wqkv^T(4096x12288) + bqkv
//   GEMM2: pool = pc(368x4096)  @ Wo^T  (4096x4096)  + bo
// Workload is HBM-bound (~268MB of fp32 weights @ 23.3 TB/s), so staging uses
// gfx1250 async global->LDS copies (ASYNCcnt) with LDS double buffering when
// the builtin is available.
// ---------------------------------------------------------------------------

typedef __attribute__((ext_vector_type(2))) float v2f;
typedef __attribute__((ext_vector_type(4))) float v4f;
typedef __attribute__((ext_vector_type(8))) float v8f;
typedef __attribute__((ext_vector_type(4))) int   v4i;

typedef __attribute__((address_space(1))) v4i glb_v4i;  // global int4
typedef __attribute__((address_space(3))) v4i lds_v4i;  // LDS int4

#define TLEN   24
#define BATCH  16
#define NHEADS 8
#define HD     512
#define EDIM   4096

// Padded LDS stride (floats): 36*4 = 144 bytes -> 16B aligned rows (b128 ok),
// and 36*r mod 64 walks distinct bank groups (conflict-free b64 frag reads).
#define LDT 36
#define KC  32

#if __has_builtin(__builtin_amdgcn_global_load_async_to_lds_b128)
#define HAVE_ASYNC_LDS 1
#else
#define HAVE_ASYNC_LDS 0
#endif

__device__ __forceinline__ void copy16_to_lds(float* lds_dst, const float* gsrc) {
#if HAVE_ASYNC_LDS
  __builtin_amdgcn_global_load_async_to_lds_b128((glb_v4i*)gsrc, (lds_v4i*)lds_dst,
                                                 0, 0);
#else
  *(v4f*)lds_dst = *(const v4f*)gsrc;
#endif
}

__device__ __forceinline__ void async_copy_wait() {
#if HAVE_ASYNC_LDS
#if __has_builtin(__builtin_amdgcn_s_wait_asynccnt)
  __builtin_amdgcn_s_wait_asynccnt(0);
#else
  asm volatile("s_wait_asynccnt 0x0" ::: "memory");
#endif
#endif
}

// ---------------------------------------------------------------------------
// Kernel 0: x (B,F,T,N) -> xt rows r=t*B+b, cols e=f*64+n  (384 x 4096)
// ---------------------------------------------------------------------------
__global__ __launch_bounds__(256) void k_transpose_x(const float* __restrict__ x,
                                                     float* __restrict__ xt) {
  int idx = blockIdx.x * 256 + threadIdx.x;           // 384*4096 = 1572864
  int n = idx & 63;
  int f = (idx >> 6) & 63;
  int b = (idx >> 12) & 15;
  int t = idx >> 16;                                   // 0..23
  xt[idx] = x[(((size_t)(b * 64 + f)) * TLEN + t) * 64 + n];
}

// ---------------------------------------------------------------------------
// Kernel 1/3: C(MxN) = A(MxK) * B(NxK)^T + bias, fp32 via WMMA 16x16x4.
// Block tile 32(M) x 256(N), K chunk 32, LDS double-buffered.
// 8 waves: wm in {0,1} (16 rows each), wn in {0..3} (64 cols each);
// each wave owns a 16x64 strip = four v8f accumulators.
// ---------------------------------------------------------------------------
__global__ __launch_bounds__(256) void k_gemm_nt_f32_wmma(
    const float* __restrict__ A, const float* __restrict__ B,
    const float* __restrict__ bias, float* __restrict__ C,
    int M, int N, int K) {
  __shared__ float sA[2][32 * LDT];
  __shared__ float sB[2][256 * LDT];

  const int tid  = threadIdx.x;
  const int lane = tid & 31;
  const int wave = tid >> 5;
  const int wm   = wave & 1;   // M sub-tile (16 rows)
  const int wn   = wave >> 1;  // N group (64 cols)

  const int mBase = blockIdx.y * 32;
  const int nBase = blockIdx.x * 256;

  // --- global staging assignments (16B per copy op) ---
  const int ldRow = tid >> 3;          // 0..31
  const int ldCol = (tid & 7) * 4;     // 0,4,...,28
  int aRow = mBase + ldRow;
  if (aRow > M - 1) aRow = M - 1;      // clamp partial M tile (dup rows, unused)
  const float* Aptr = A + (size_t)aRow * K + ldCol;
  const float* Bptr = B + (size_t)(nBase + ldRow) * K + ldCol;

  // --- WMMA fragment addressing (per ISA 16x16x4 f32 layouts) ---
  const int lm = lane & 15;            // row/col within 16
  const int kp = (lane >> 4) * 2;      // K pair select

  v8f acc[4];
#pragma unroll
  for (int s = 0; s < 4; ++s) acc[s] = (v8f){0.f, 0.f, 0.f, 0.f, 0.f, 0.f, 0.f, 0.f};

  // stage chunk k0 into buffer `buf`
  auto stage = [&](int k0, int buf) {
    copy16_to_lds(&sA[buf][ldRow * LDT + ldCol], Aptr + k0);
#pragma unroll
    for (int i = 0; i < 8; ++i) {
      copy16_to_lds(&sB[buf][(ldRow + 32 * i) * LDT + ldCol],
                    Bptr + (size_t)(32 * i) * K + k0);
    }
  };

  stage(0, 0);
  async_copy_wait();
  __syncthreads();

  int buf = 0;
  for (int k0 = 0; k0 < K; k0 += KC) {
    if (k0 + KC < K) stage(k0 + KC, buf ^ 1);   // overlap next-chunk staging
    if (k0 + 2 * KC < K) {                      // speculative deep prefetch
      __builtin_prefetch(Aptr + k0 + 2 * KC, 0, 0);
      __builtin_prefetch(Bptr + k0 + 2 * KC, 0, 0);
    }

#pragma unroll
    for (int kk = 0; kk < KC; kk += 4) {
      v2f af = *(const v2f*)&sA[buf][(wm * 16 + lm) * LDT + kk + kp];
#pragma unroll
      for (int s = 0; s < 4; ++s) {
        v2f bf = *(const v2f*)&sB[buf][(wn * 64 + s * 16 + lm) * LDT + kk + kp];
        acc[s] = __builtin_amdgcn_wmma_f32_16x16x4_f32(false, af, false, bf,
                                                       (short)0, acc[s],
                                                       false, false);
      }
    }

    async_copy_wait();    // own async copies into buf^1 complete
    __syncthreads();      // everyone's copies + reads of buf complete
    buf ^= 1;
  }

  // --- store D (+bias): VGPR i -> row wm*16 + (lane>=16 ? 8:0) + i, col lane&15
  const int rBase = mBase + wm * 16 + ((lane >> 4) << 3);
  const int c0 = nBase + wn * 64 + lm;
#pragma unroll
  for (int s = 0; s < 4; ++s) {
    const float bv = bias[c0 + s * 16];
#pragma unroll
    for (int i = 0; i < 8; ++i) {
      int r = rBase + i;
      if (r < M) C[(size_t)r * N + c0 + s * 16] = acc[s][i] + bv;
    }
  }
}

// ---------------------------------------------------------------------------
// Kernel 2: per-(b,h) attention + query-pool.
// pc[t*B+b, h*512+d] = (1/(t+1)) * sum_{q<=t} sum_{k<=t} softmax_{k<=t}(S[q,:])[k] * V[k,d]
// with S[q,k] = (Q[q].K[k])/sqrt(512) + (k<=q ? 1 : 0).
// ---------------------------------------------------------------------------
__global__ __launch_bounds__(256) void k_attn_pool(const float* __restrict__ qkv,
                                                   float* __restrict__ pc) {
  const int b = blockIdx.x >> 3;
  const int h = blockIdx.x & 7;

  __shared__ float sQ[TLEN * HD];   // 48KB, reused for V after S is built
  __shared__ float sK[TLEN * HD];   // 48KB
  __shared__ float sS[TLEN][TLEN];
  __shared__ float sP[TLEN][TLEN];
  __shared__ float sW[TLEN];

  const int tid = threadIdx.x;

  // stage Q and K
  for (int idx = tid; idx < TLEN * HD; idx += 256) {
    int t = idx >> 9, d = idx & 511;
    size_t base = (size_t)(t * BATCH + b) * (3 * EDIM) + (size_t)h * HD + d;
    sQ[idx] = qkv[base];
    sK[idx] = qkv[base + EDIM];
  }
  __syncthreads();

  const float scale = 0.044194173824159216f;  // 1/sqrt(512)
  for (int p = tid; p < TLEN * TLEN; p += 256) {
    int q = p / TLEN, kk = p % TLEN;
    const float* qr = &sQ[q * HD];
    const float* kr = &sK[kk * HD];
    float acc = 0.f;
#pragma unroll 4
    for (int d = 0; d < HD; ++d) acc = fmaf(qr[d], kr[d], acc);
    sS[q][kk] = acc * scale + (kk <= q ? 1.0f : 0.0f);
  }
  __syncthreads();

  // stage V over sQ
  for (int idx = tid; idx < TLEN * HD; idx += 256) {
    int t = idx >> 9, d = idx & 511;
    sQ[idx] = qkv[(size_t)(t * BATCH + b) * (3 * EDIM) + 2 * EDIM + (size_t)h * HD + d];
  }
  __syncthreads();

  for (int t = 0; t < TLEN - 1; ++t) {
    if (tid < TLEN) {  // softmax of row tid over k<=t
      int q = tid;
      float m = -1e30f;
      for (int k = 0; k <= t; ++k) m = fmaxf(m, sS[q][k]);
      float s = 0.f;
      for (int k = 0; k <= t; ++k) {
        float e = __expf(sS[q][k] - m);
        sP[q][k] = e;
        s += e;
      }
      float inv = 1.0f / s;
      for (int k = 0; k <= t; ++k) sP[q][k] *= inv;
    }
    __syncthreads();
    if (tid <= t) {    // column sum over q<=t, fold 1/(t+1)
      float s = 0.f;
      for (int q = 0; q <= t; ++q) s += sP[q][tid];
      sW[tid] = s / (float)(t + 1);
    }
    __syncthreads();
    for (int d = tid; d < HD; d += 256) {
      float acc = 0.f;
      for (int k = 0; k <= t; ++k) acc = fmaf(sW[k], sQ[k * HD + d], acc);
      pc[(size_t)(t * BATCH + b) * EDIM + (size_t)h * HD + d] = acc;
    }
    __syncthreads();
  }
}

// ---------------------------------------------------------------------------
// Kernel 4: pred = relu(pool.reshape(23*B*64, 64) @ Wfc^T + bfc), scattered to
// out[b, i, 1+t, j]. Block = 4 rows x 64 cols; Wfc cached in LDS.
// ---------------------------------------------------------------------------
__global__ __launch_bounds__(256) void k_fc_relu_scatter(
    const float* __restrict__ pool, const float* __restrict__ Wfc,
    const float* __restrict__ bfc, float* __restrict__ out) {
  __shared__ float sW[64 * 65];
  __shared__ float sIn[4 * 64];
  const int tid = threadIdx.x;

  for (int i = tid; i < 64 * 64; i += 256) sW[(i >> 6) * 65 + (i & 63)] = Wfc[i];

  const int rl = tid >> 6;  // 0..3 local row
  const int j  = tid & 63;
  const int rr = blockIdx.x * 4 + rl;   // 0..23551
  const int i  = rr & 63;
  const int tb = rr >> 6;
  const int b  = tb & 15;
  const int t  = tb >> 4;               // 0..22

  sIn[rl * 64 + j] = pool[(size_t)(t * BATCH + b) * EDIM + i * 64 + j];
  __syncthreads();

  float acc = bfc[j];
  const float* wrow = &sW[j * 65];
  const float* inr  = &sIn[rl * 64];
#pragma unroll 8
  for (int c = 0; c < 64; ++c) acc = fmaf(inr[c], wrow[c], acc);
  acc = fmaxf(acc, 0.f);
  out[((size_t)(b * 64 + i) * TLEN + (1 + t)) * 64 + j] = acc;
}

// ---------------------------------------------------------------------------
// Kernel 5: out[:,:,0,:] = x[:,:,0,:]  (identical flat layout)
// ---------------------------------------------------------------------------
__global__ __launch_bounds__(256) void k_copy_first(const float* __restrict__ x,
                                                    float* __restrict__ out) {
  int idx = blockIdx.x * 256 + threadIdx.x;  // 65536 = B*F*N
  int n = idx & 63;
  int bf = idx >> 6;
  size_t off = ((size_t)bf * TLEN) * 64 + n;
  out[off] = x[off];
}

// ---------------------------------------------------------------------------
extern "C" void kernel_launch(void* const* d_in, const int* in_sizes, int n_in,
                              void* d_out, int out_size, void* d_ws, size_t ws_size,
                              hipStream_t stream) {
  const float* x    = (const float*)d_in[0];
  const float* Wqkv = (const float*)d_in[1];
  const float* bqkv = (const float*)d_in[2];
  const float* Wo   = (const float*)d_in[3];
  const float* bo   = (const float*)d_in[4];
  const float* Wfc  = (const float*)d_in[5];
  const float* bfc  = (const float*)d_in[6];
  float* out = (float*)d_out;

  char* ws = (char*)d_ws;
  // xt (6.29MB) is dead after GEMM1 -> alias pool over it.
  float* xt   = (float*)(ws);                       //   384*4096*4 = 6291456
  float* pool = (float*)(ws);                       //   alias of xt
  float* qkv  = (float*)(ws + 6291456);             //   384*12288*4 = 18874368
  float* pc   = (float*)(ws + 6291456 + 18874368);  //   384*4096*4 = 6291456
  // total ws use: 31457280 bytes

  // 0) transpose x -> xt
  k_transpose_x<<<(384 * 4096) / 256, 256, 0, stream>>>(x, xt);

  // 1) qkv = xt @ Wqkv^T + bqkv   (M=384, N=12288, K=4096)
  {
    dim3 grid(12288 / 256, 384 / 32);
    k_gemm_nt_f32_wmma<<<grid, 256, 0, stream>>>(xt, Wqkv, bqkv, qkv,
                                                 384, 12288, 4096);
  }

  // 2) attention + pool -> pc (368 x 4096)
  k_attn_pool<<<BATCH * NHEADS, 256, 0, stream>>>(qkv, pc);

  // 3) pool = pc @ Wo^T + bo      (M=368, N=4096, K=4096)
  {
    dim3 grid(4096 / 256, (368 + 31) / 32);
    k_gemm_nt_f32_wmma<<<grid, 256, 0, stream>>>(pc, Wo, bo, pool,
                                                 368, 4096, 4096);
  }

  // 4) FC + ReLU + scatter to out[:, :, 1:, :]
  k_fc_relu_scatter<<<23552 / 4, 256, 0, stream>>>(pool, Wfc, bfc, out);

  // 5) out[:, :, 0, :] = x[:, :, 0, :]
  k_copy_first<<<65536 / 256, 256, 0, stream>>>(x, out);
}